// CompleteGenuineSFT_39986145526128
// MI455X (gfx1250) — compile-verified
//
#include <hip/hip_runtime.h>
#include <math.h>

typedef float v8f __attribute__((ext_vector_type(8)));
typedef float v2f __attribute__((ext_vector_type(2)));

#define B_   16
#define S_   2048
#define D_   256
#define P_   512

__device__ __forceinline__ float geluf(float x) {
    return 0.5f * x * (1.0f + erff(x * 0.70710678118654752440f));
}
__device__ __forceinline__ float sigmoidf_(float x) {
    return 1.0f / (1.0f + expf(-x));
}

// ---------------------------------------------------------------------------
// f32 WMMA GEMM, K contiguous in both operands:
//   C[m,n] = act( sum_sh sum_k A[m+shiftBase+sh, k] * B[sh][k,n] + bias[n] ) * rowScale[m]
// A element (m,k) at A + m*lda_m + k ; B element (k,n) at B + sh*bShiftStride + n*ldb_n + k
// One 16x16 tile per wave, 4 waves per block. K%4==0, N%16==0, M%16==0.
// GUARD: clamp+mask A rows that fall outside [0,Arows) (for shifted conv GEMMs).
// ---------------------------------------------------------------------------
template<int ACT, bool ABS, bool BIAS, bool RSCALE, bool GUARD>
__global__ __launch_bounds__(128) void gemm_wmma(
    const float* __restrict__ A, const float* __restrict__ B, float* __restrict__ C,
    int M, int N, int K, int Arows,
    long lda_m, long ldb_n, long ldc,
    int nshift, int shiftBase, long bShiftStride,
    const float* __restrict__ bias, const float* __restrict__ rowScale,
    long aBatch, long bBatch, long cBatch)
{
    const int wave  = threadIdx.x >> 5;
    const int lane  = threadIdx.x & 31;
    const int half  = lane >> 4;        // 0: K pair {0,1}, 1: K pair {2,3}
    const int lm    = lane & 15;
    const int tileN = (blockIdx.x * 4 + wave) * 16;
    const int tileM = blockIdx.y * 16;
    if (tileN >= N) return;

    const long bz = blockIdx.z;
    const float* Ab = A + bz * aBatch;
    const float* Bb = B + bz * bBatch;
    float*       Cb = C + bz * cBatch;

    const int mRow = tileM + lm;
    const int nCol = tileN + lm;
    const float* bcolBase = Bb + (long)nCol * ldb_n + half * 2;

    v8f acc = {0.f, 0.f, 0.f, 0.f, 0.f, 0.f, 0.f, 0.f};

    for (int sh = 0; sh < nshift; ++sh) {
        const int r = mRow + shiftBase + sh;
        float mask = 1.0f;
        int rc = r;
        if (GUARD) {
            mask = (r >= 0 && r < Arows) ? 1.0f : 0.0f;
            rc = r < 0 ? 0 : (r >= Arows ? Arows - 1 : r);
        }
        const float* ap = Ab + (long)rc * lda_m + half * 2;
        const float* bp = bcolBase + (long)sh * bShiftStride;
        #pragma unroll 2
        for (int kk = 0; kk < K; kk += 4) {
            v2f av = *(const v2f*)(ap + kk);
            v2f bv = *(const v2f*)(bp + kk);
            if (GUARD) av *= mask;
            if (ABS) { av[0] = fabsf(av[0]); av[1] = fabsf(av[1]); }
            acc = __builtin_amdgcn_wmma_f32_16x16x4_f32(
                      false, av, false, bv, (short)0, acc, false, false);
        }
    }

    #pragma unroll
    for (int i = 0; i < 8; ++i) {
        const int m = tileM + i + 8 * half;
        const int n = nCol;
        if (m < M && n < N) {
            float v = acc[i];
            if (BIAS)     v += bias[n];
            if (ACT == 1) v = geluf(v);
            if (ACT == 2) v = tanhf(v);
            if (RSCALE)   v *= rowScale[m];
            Cb[(long)m * ldc + n] = v;
        }
    }
}

// weight repack: out[sh][n][k] = in[n][k][sh]   (conv weights [Cout,Cin,KS] -> per-shift [Cout,Cin])
__global__ __launch_bounds__(256) void k_packw(const float* __restrict__ in, float* __restrict__ out,
                                               int N, int K, int NS)
{
    long idx = (long)blockIdx.x * 256 + threadIdx.x;    // ((sh*N)+n)*K + k
    long tot = (long)N * K * NS;
    if (idx >= tot) return;
    int k = (int)(idx % K);
    long t = idx / K;
    int n = (int)(t % N);
    int sh = (int)(t / N);
    out[idx] = in[((long)n * K + k) * NS + sh];
}

// interference factor[p] = 1 + 0.1*tanh( sum_i amp_i * sin(2*pi*f_i*pos_p) )
__global__ __launch_bounds__(512) void k_wave(const float* __restrict__ freq,
                                              const float* __restrict__ amp,
                                              float* __restrict__ factor)
{
    int p = threadIdx.x;
    if (p >= P_) return;
    float pos = (float)p / (float)(P_ - 1);
    float w = 0.f;
    #pragma unroll
    for (int i = 0; i < 4; ++i)
        w += amp[i] * sinf(6.2831853071795864769f * freq[i] * pos);
    factor[p] = 1.0f + 0.1f * tanhf(w);
}

// Hs[l] = 0.5*(pot[l] + pot[l]^T)
__global__ __launch_bounds__(256) void k_hs(const float* __restrict__ pot, float* __restrict__ Hs)
{
    int idx = blockIdx.x * 256 + threadIdx.x;           // l*65536 + r*256 + c
    if (idx >= 4 * D_ * D_) return;
    int l = idx >> 16, rc = idx & 65535, r = rc >> 8, c = rc & 255;
    Hs[idx] = 0.5f * (pot[l * 65536 + r * 256 + c] + pot[l * 65536 + c * 256 + r]);
}

// h1[b,s,c] = gelu( cb1[c] + sum_k cw1[c, byte[b,s+k-3], k] )   (conv1 on one-hot == gather)
__global__ __launch_bounds__(256) void k_h1(const int* __restrict__ bytes,
                                            const float* __restrict__ cw1,
                                            const float* __restrict__ cb1,
                                            float* __restrict__ h1)
{
    long idx = (long)blockIdx.x * 256 + threadIdx.x;    // (b*S + s)*128 + c
    if (idx >= (long)B_ * S_ * 128) return;
    int c  = (int)(idx & 127);
    long bs = idx >> 7;
    int s  = (int)(bs & (S_ - 1));
    long b = bs >> 11;
    float acc = cb1[c];
    const float* wrow = cw1 + (long)c * 256 * 7;
    #pragma unroll
    for (int k = 0; k < 7; ++k) {
        int sp = s + k - 3;
        if (sp >= 0 && sp < S_) {
            int by = bytes[b * S_ + sp];
            acc += wrow[by * 7 + k];
        }
    }
    h1[idx] = geluf(acc);
}

// ctx[b,s] = sigmoid( cb3 + sum_{c,k} cw3[c,k] * h2[b, s+k-1, c] )
__global__ __launch_bounds__(256) void k_ctx(const float* __restrict__ h2,
                                             const float* __restrict__ cw3,
                                             const float* __restrict__ cb3,
                                             float* __restrict__ ctx)
{
    long idx = (long)blockIdx.x * 256 + threadIdx.x;    // b*S + s
    if (idx >= (long)B_ * S_) return;
    int s  = (int)(idx & (S_ - 1));
    long b = idx >> 11;
    float acc = cb3[0];
    #pragma unroll
    for (int k = 0; k < 3; ++k) {
        int sp = s + k - 1;
        if (sp >= 0 && sp < S_) {
            const float* hrow = h2 + (b * S_ + sp) * 64;
            for (int c = 0; c < 64; ++c) acc += cw3[c * 3 + k] * hrow[c];
        }
    }
    ctx[idx] = sigmoidf_(acc);
}

// per-(b,s) wave-packet params: center, wbase, ascale, phase
__global__ __launch_bounds__(256) void k_locp(const int* __restrict__ bytes,
                                              const float* __restrict__ loc_shapes,
                                              float* __restrict__ locp)
{
    long idx = (long)blockIdx.x * 256 + threadIdx.x;    // b*S + s
    if (idx >= (long)B_ * S_) return;
    int s = (int)(idx & (S_ - 1));
    const float* lp = loc_shapes + (long)bytes[idx] * 4;
    float* o = locp + idx * 4;
    o[0] = (float)s / (float)(S_ - 1) + lp[0] * 0.05f;  // center
    o[1] = fabsf(lp[1]);                                // wbase
    o[2] = sigmoidf_(lp[2]);                            // ascale
    o[3] = lp[3] * 3.14159265358979323846f;             // phase
}

// mod_amp transposed: modampT[b][d][s] = emb[byte[b,s], d] * ctx[b,s]
__global__ __launch_bounds__(256) void k_modampT(const int* __restrict__ bytes,
                                                 const float* __restrict__ emb,
                                                 const float* __restrict__ ctx,
                                                 float* __restrict__ modampT)
{
    long idx = (long)blockIdx.x * 256 + threadIdx.x;    // ((b*D)+d)*S + s
    if (idx >= (long)B_ * D_ * S_) return;
    int s = (int)(idx & (S_ - 1));
    long t = idx >> 11;
    int d = (int)(t & 255);
    long b = t >> 8;
    long bs = b * S_ + s;
    modampT[idx] = emb[(long)bytes[bs] * D_ + d] * ctx[bs];
}

// basis[p][s] for one batch: sum over 4 scales of Gaussian*cos wave packets
__global__ __launch_bounds__(256) void k_basis(const float* __restrict__ locp,   // offset to batch
                                               const float* __restrict__ msw,
                                               const float* __restrict__ sw,
                                               float* __restrict__ basis)
{
    long idx = (long)blockIdx.x * 256 + threadIdx.x;    // p*S + s
    if (idx >= (long)P_ * S_) return;
    int s = (int)(idx & (S_ - 1));
    int p = (int)(idx >> 11);
    const float* lp = locp + (long)s * 4;
    float center = lp[0], wbase = lp[1], asc = lp[2], phase = lp[3];
    float xd = (float)p / (float)(P_ - 1) - center;
    float total = 0.f;
    #pragma unroll
    for (int k = 0; k < 4; ++k) {
        float width = wbase * msw[k] + 1e-5f;
        float z = xd / width;
        float g = expf(-0.5f * z * z) * cosf(phase + 6.2831853071795864769f * z);
        total += g * (asc * sw[k] * sw[k]);
    }
    basis[idx] = total;
}

__global__ void k_zero16(float* p) { if (threadIdx.x < 16) p[threadIdx.x] = 0.f; }

// per-batch norm^2 accumulation: sum over (p,d) of psi^2 + im^2
__global__ __launch_bounds__(256) void k_hred(const float* __restrict__ cur,
                                              const float* __restrict__ Gl,
                                              const float* __restrict__ NL,
                                              const float* __restrict__ kinp,
                                              const float* __restrict__ evtp,
                                              float* __restrict__ normsq)
{
    __shared__ float sm[256];
    int bp = blockIdx.x;                 // b*512 + p
    int b = bp >> 9, p = bp & (P_ - 1);
    int d = threadIdx.x;
    long base = (long)bp * D_;
    float psi = cur[base + d];
    int pp = (p == P_ - 1) ? 0 : p + 1;
    int pm = (p == 0) ? P_ - 1 : p - 1;
    float lap = (cur[((long)(b * P_ + pp)) * D_ + d] - 2.f * psi
               + cur[((long)(b * P_ + pm)) * D_ + d]) * (float)(P_ * P_);
    float H = -kinp[0] * lap + Gl[base + d] + 0.1f * NL[base + d] * psi;
    float im = -evtp[0] * H;
    float v = psi * psi + im * im;
    sm[d] = v; __syncthreads();
    for (int s = 128; s > 0; s >>= 1) { if (d < s) sm[d] += sm[d + s]; __syncthreads(); }
    if (d == 0) atomicAdd(&normsq[b], sm[0]);
}

// cur = rownorm( cur * (1 + 1/(sqrt(normsq_b)+1e-8)) )
__global__ __launch_bounds__(256) void k_update(float* __restrict__ cur,
                                                const float* __restrict__ normsq)
{
    __shared__ float sm[256];
    int bp = blockIdx.x;
    int b = bp >> 9;
    int d = threadIdx.x;
    long base = (long)bp * D_;
    float nb = sqrtf(normsq[b]);
    float c = cur[base + d] * (1.0f + 1.0f / (nb + 1e-8f));
    sm[d] = c * c; __syncthreads();
    for (int s = 128; s > 0; s >>= 1) { if (d < s) sm[d] += sm[d + s]; __syncthreads(); }
    float rn = sqrtf(sm[0]);
    __syncthreads();
    cur[base + d] = c / (rn + 1e-8f);
}

// GroupNorm stats over (chPer channels, all P): one block per (b,g)
__global__ __launch_bounds__(256) void k_gnstat(const float* __restrict__ x, int C, int chPer,
                                                float* __restrict__ st)
{
    __shared__ float s0[256], s1[256];
    int bg = blockIdx.x;
    int b = bg >> 3, g = bg & 7;
    int n = chPer * P_;
    float sum = 0.f, sq = 0.f;
    for (int i = threadIdx.x; i < n; i += 256) {
        int c = i % chPer, p = i / chPer;
        float v = x[((long)(b * P_ + p)) * C + g * chPer + c];
        sum += v; sq += v * v;
    }
    s0[threadIdx.x] = sum; s1[threadIdx.x] = sq; __syncthreads();
    for (int s = 128; s > 0; s >>= 1) {
        if (threadIdx.x < s) { s0[threadIdx.x] += s0[threadIdx.x + s]; s1[threadIdx.x] += s1[threadIdx.x + s]; }
        __syncthreads();
    }
    if (threadIdx.x == 0) {
        float mu = s0[0] / (float)n;
        st[bg * 2 + 0] = mu;
        st[bg * 2 + 1] = s1[0] / (float)n - mu * mu;
    }
}

__global__ __launch_bounds__(256) void k_gnapply(float* __restrict__ x, int C, int chPer,
                                                 const float* __restrict__ st,
                                                 const float* __restrict__ gamma,
                                                 const float* __restrict__ beta)
{
    long idx = (long)blockIdx.x * 256 + threadIdx.x;    // (b*P+p)*C + c
    if (idx >= (long)B_ * P_ * C) return;
    int c = (int)(idx % C);
    long bp = idx / C;
    int b = (int)(bp >> 9);
    int g = c / chPer;
    float mu = st[(b * 8 + g) * 2], var = st[(b * 8 + g) * 2 + 1];
    float y = (x[idx] - mu) * rsqrtf(var + 1e-5f) * gamma[c] + beta[c];
    x[idx] = geluf(y);
}

// attention logits + softmax over P (one block per batch)
__global__ __launch_bounds__(512) void k_att(const float* __restrict__ h2,
                                             const float* __restrict__ rw3,
                                             const float* __restrict__ rb3,
                                             float* __restrict__ attw)
{
    __shared__ float sm[512];
    int b = blockIdx.x, p = threadIdx.x;
    const float* row = h2 + ((long)(b * P_ + p)) * D_;
    float acc = rb3[0];
    for (int c = 0; c < D_; ++c) acc += rw3[c] * row[c];
    sm[p] = acc; __syncthreads();
    for (int s = 256; s > 0; s >>= 1) { if (p < s) sm[p] = fmaxf(sm[p], sm[p + s]); __syncthreads(); }
    float mx = sm[0]; __syncthreads();
    float e = expf(acc - mx);
    sm[p] = e; __syncthreads();
    for (int s = 256; s > 0; s >>= 1) { if (p < s) sm[p] += sm[p + s]; __syncthreads(); }
    attw[(long)b * P_ + p] = e / sm[0];
}

__global__ __launch_bounds__(256) void k_pool(const float* __restrict__ attw,
                                              const float* __restrict__ cur,
                                              float* __restrict__ pooled)
{
    int b = blockIdx.x, d = threadIdx.x;
    float acc = 0.f;
    for (int p = 0; p < P_; ++p)
        acc += attw[(long)b * P_ + p] * cur[((long)(b * P_ + p)) * D_ + d];
    pooled[(long)b * D_ + d] = acc;
}

__global__ __launch_bounds__(128) void k_clf(const float* __restrict__ pooled,
                                             const float* __restrict__ w1, const float* __restrict__ b1,
                                             const float* __restrict__ w2, const float* __restrict__ b2,
                                             float* __restrict__ logits)
{
    __shared__ float hid[128];
    int b = blockIdx.x, h = threadIdx.x;
    const float* pb = pooled + (long)b * D_;
    float acc = b1[h];
    for (int d = 0; d < D_; ++d) acc += w1[(long)h * D_ + d] * pb[d];
    hid[h] = geluf(acc);
    __syncthreads();
    if (h < 2) {
        float l = b2[h];
        for (int j = 0; j < 128; ++j) l += w2[(long)h * 128 + j] * hid[j];
        logits[b * 2 + h] = l;
    }
}

// ---------------------------------------------------------------------------
extern "C" void kernel_launch(void* const* d_in, const int* in_sizes, int n_in,
                              void* d_out, int out_size, void* d_ws, size_t ws_size,
                              hipStream_t stream)
{
    (void)in_sizes; (void)n_in; (void)out_size; (void)ws_size;
    const int*   bytes   = (const int*)  d_in[0];
    const float* emb     = (const float*)d_in[1];
    const float* locsh   = (const float*)d_in[2];
    const float* cw1     = (const float*)d_in[3];
    const float* cb1     = (const float*)d_in[4];
    const float* cw2     = (const float*)d_in[5];
    const float* cb2     = (const float*)d_in[6];
    const float* cw3     = (const float*)d_in[7];
    const float* cb3     = (const float*)d_in[8];
    const float* msw     = (const float*)d_in[9];
    const float* sw      = (const float*)d_in[10];
    const float* ifreq   = (const float*)d_in[11];
    const float* iamp    = (const float*)d_in[12];
    const float* pot     = (const float*)d_in[13];
    const float* kin     = (const float*)d_in[14];
    const float* nw1     = (const float*)d_in[15];
    const float* nb1     = (const float*)d_in[16];
    const float* nw2     = (const float*)d_in[17];
    const float* nb2     = (const float*)d_in[18];
    const float* evt     = (const float*)d_in[19];
    const float* rw1     = (const float*)d_in[20];
    const float* rb1     = (const float*)d_in[21];
    const float* g1g     = (const float*)d_in[22];
    const float* g1b     = (const float*)d_in[23];
    const float* rw2     = (const float*)d_in[24];
    const float* rb2     = (const float*)d_in[25];
    const float* g2g     = (const float*)d_in[26];
    const float* g2b     = (const float*)d_in[27];
    const float* rw3     = (const float*)d_in[28];
    const float* rb3     = (const float*)d_in[29];
    const float* clfw1   = (const float*)d_in[30];
    const float* clfb1   = (const float*)d_in[31];
    const float* clfw2   = (const float*)d_in[32];
    const float* clfb2   = (const float*)d_in[33];

    float* out    = (float*)d_out;
    float* logits = out;            // [16,2]
    float* cur    = out + 32;       // [16,512,256]

    float* ws = (float*)d_ws;       // ~69 MB arena with phase reuse
    float* modampT = ws;                        // 8,388,608  [b][d][s]  (phase A/field)
    float* Gl     = ws;                         // 2,097,152  (phase B reuse)
    float* NL     = ws + 2097152;               // 2,097,152
    float* Tbuf   = ws + 4194304;               // 1,048,576
    float* Hr1    = ws;                         // 4,194,304  (phase C reuse)
    float* Hr2    = ws + 4194304;               // 2,097,152
    float* h1     = ws + 8388608;               // 4,194,304
    float* h2     = ws + 12582912;              // 2,097,152
    float* ctx    = ws + 14680064;              // 32,768
    float* locp   = ws + 14712832;              // 131,072
    float* basis  = ws + 14843904;              // 1,048,576 [p][s] (per batch)
    float* factor = ws + 15892480;              // 512
    float* Hs     = ws + 15892992;              // 262,144
    float* normsq = ws + 16155136;              // 16
    float* gnst   = ws + 16155152;              // 256
    float* attw   = ws + 16155408;              // 8,192
    float* pooled = ws + 16163600;              // 4,096
    float* w2t    = ws + 16167696;              // 40,960   [5][64][128]
    float* rw1t   = ws + 16208656;              // 655,360  [5][512][256]
    float* rw2t   = ws + 16864016;              // 393,216  [3][256][512]

    // ---- small precomputes / weight repacks ----
    k_wave<<<1, 512, 0, stream>>>(ifreq, iamp, factor);
    k_hs<<<1024, 256, 0, stream>>>(pot, Hs);
    k_locp<<<128, 256, 0, stream>>>(bytes, locsh, locp);
    k_packw<<<160, 256, 0, stream>>>(cw2, w2t, 64, 128, 5);
    k_packw<<<2560, 256, 0, stream>>>(rw1, rw1t, 512, 256, 5);
    k_packw<<<1536, 256, 0, stream>>>(rw2, rw2t, 256, 512, 3);

    // ---- local semantic field context net ----
    k_h1<<<16384, 256, 0, stream>>>(bytes, cw1, cb1, h1);
    // conv2 as 5-shift WMMA GEMM: h2[b,s,c2] = gelu(cb2 + sum h1[b,s+k-2,c1]*w2t[k][c2][c1])
    gemm_wmma<1, false, true, false, true><<<dim3(1, 128, 16), 128, 0, stream>>>(
        h1, w2t, h2, S_, 64, 128, S_,
        128, 128, 64,
        5, -2, 64 * 128, cb2, nullptr,
        (long)S_ * 128, 0, (long)S_ * 64);
    k_ctx<<<128, 256, 0, stream>>>(h2, cw3, cb3, ctx);
    k_modampT<<<32768, 256, 0, stream>>>(bytes, emb, ctx, modampT);

    // ---- field synthesis: per batch, basis[p][s] then field[b] = basis @ modampT[b]^T, * factor[p]
    for (int b = 0; b < B_; ++b) {
        k_basis<<<4096, 256, 0, stream>>>(locp + (long)b * S_ * 4, msw, sw, basis);
        gemm_wmma<0, false, false, true, false><<<dim3(4, 32, 1), 128, 0, stream>>>(
            basis, modampT + (long)b * D_ * S_, cur + (long)b * P_ * D_,
            P_, D_, S_, P_,
            S_, S_, D_,
            1, 0, 0, nullptr, factor,
            0, 0, 0);
    }

    // ---- 4 Schroedinger layers ----
    for (int i = 0; i < 4; ++i) {
        // linear: Gl = psi @ Hs_i^T  (Hs row-major [n][k])
        gemm_wmma<0, false, false, false, false><<<dim3(4, 32, 16), 128, 0, stream>>>(
            cur, Hs + (long)i * D_ * D_, Gl, P_, D_, D_, P_,
            D_, D_, D_,
            1, 0, 0, nullptr, nullptr,
            (long)P_ * D_, 0, (long)P_ * D_);
        // T = tanh(|psi| @ nw1_i^T + nb1_i)   (nw1 row-major [n=128][k=256])
        gemm_wmma<2, true, true, false, false><<<dim3(2, 32, 16), 128, 0, stream>>>(
            cur, nw1 + (long)i * 128 * D_, Tbuf, P_, 128, D_, P_,
            D_, D_, 128,
            1, 0, 0, nb1 + (long)i * 128, nullptr,
            (long)P_ * D_, 0, (long)P_ * 128);
        // NL = T @ nw2_i^T + nb2_i            (nw2 row-major [n=256][k=128])
        gemm_wmma<0, false, true, false, false><<<dim3(4, 32, 16), 128, 0, stream>>>(
            Tbuf, nw2 + (long)i * D_ * 128, NL, P_, D_, 128, P_,
            128, 128, D_,
            1, 0, 0, nb2 + (long)i * D_, nullptr,
            (long)P_ * 128, 0, (long)P_ * D_);
        k_zero16<<<1, 32, 0, stream>>>(normsq);
        k_hred<<<B_ * P_, 256, 0, stream>>>(cur, Gl, NL, kin + i, evt + i, normsq);
        k_update<<<B_ * P_, 256, 0, stream>>>(cur, normsq);
    }

    // ---- rich field processor ----
    // conv rw1 (K=5, pad 2): Hr1[b,p,co(512)]
    gemm_wmma<0, false, true, false, true><<<dim3(8, 32, 16), 128, 0, stream>>>(
        cur, rw1t, Hr1, P_, 512, D_, P_,
        D_, D_, 512,
        5, -2, 512 * 256, rb1, nullptr,
        (long)P_ * D_, 0, (long)P_ * 512);
    k_gnstat<<<128, 256, 0, stream>>>(Hr1, 512, 64, gnst);
    k_gnapply<<<16384, 256, 0, stream>>>(Hr1, 512, 64, gnst, g1g, g1b);
    // conv rw2 (K=3, pad 1): Hr2[b,p,co(256)]
    gemm_wmma<0, false, true, false, true><<<dim3(4, 32, 16), 128, 0, stream>>>(
        Hr1, rw2t, Hr2, P_, D_, 512, P_,
        512, 512, D_,
        3, -1, 256 * 512, rb2, nullptr,
        (long)P_ * 512, 0, (long)P_ * D_);
    k_gnstat<<<128, 256, 0, stream>>>(Hr2, 256, 32, gnst);
    k_gnapply<<<8192, 256, 0, stream>>>(Hr2, 256, 32, gnst, g2g, g2b);

    // ---- attention pooling + classifier ----
    k_att<<<B_, 512, 0, stream>>>(Hr2, rw3, rb3, attw);
    k_pool<<<B_, 256, 0, stream>>>(attw, cur, pooled);
    k_clf<<<B_, 128, 0, stream>>>(pooled, clfw1, clfb1, clfw2, clfb2, logits);
}